// RNN_base_52810917872232
// MI455X (gfx1250) — compile-verified
//
#include <hip/hip_runtime.h>
#include <math.h>

// ---------------------------------------------------------------------------
// RNN forward for MI455X (gfx1250), fp32 via V_WMMA_F32_16X16X4_F32.
//   X[512,64,512], W_xh[512,1024], W_hh[1024,1024], b_h[1024],
//   W_hq[1024,512], b_q[512]
//   outputs[512,64,512] ++ states[512,64,1024]  (concatenated in d_out)
// ---------------------------------------------------------------------------

typedef __attribute__((ext_vector_type(2))) float v2f;
typedef __attribute__((ext_vector_type(8))) float v8f;

#define WMMA_F32(a, b, c) \
  __builtin_amdgcn_wmma_f32_16x16x4_f32(false, (a), false, (b), (short)0, (c), false, false)

// CDNA5 has a hardware V_TANH_F32 transcendental; use it if clang declares it.
#if __has_builtin(__builtin_amdgcn_tanhf)
#define TANHF(x) __builtin_amdgcn_tanhf(x)
#else
#define TANHF(x) tanhf(x)
#endif

#define T_STEPS 512
#define BATCH   64
#define HID     1024
#define OUTD    512
#define IND     512
#define NWG_SCAN 64   // persistent workgroups in the scan (one 16-col slice each)

// ---------------------------------------------------------------------------
// Generic fp32 WMMA GEMM + bias:  C[M,N] = A[M,K] @ W[K,N] + bias[N]
// Block = 256 threads (8 waves). Per-block tile: 128 rows x 64 cols.
// K streamed in 64-chunks through LDS (padded to 80 words/row so the two
// lane-halves of a B-fragment load hit disjoint bank groups). The next W tile
// is prefetched into registers while computing on the current LDS tile.
// ---------------------------------------------------------------------------
template <int K, int N>
__global__ __launch_bounds__(256) void gemm_bias_kernel(
    const float* __restrict__ A, const float* __restrict__ W,
    const float* __restrict__ bias, float* __restrict__ C) {
  constexpr int KC = 64;
  constexpr int LDSW = 80;  // 64 + 16 pad
  constexpr int NCHUNK = K / KC;
  __shared__ float sW[KC * LDSW];

  const int tid = threadIdx.x;
  const int wv = tid >> 5;          // wave 0..7 -> M sub-tile
  const int ln = tid & 31;
  const int lh = ln >> 4;           // lane half (0/1)
  const int lm = ln & 15;
  const int m0 = blockIdx.x * 128 + wv * 16;
  const int n0 = blockIdx.y * 64;

  // this thread's 16 staging elements of each 64x64 W tile
  const int skk = tid >> 6;         // row 0..3 (+4 per j)
  const int snn = tid & 63;         // col 0..63

  v8f acc[4] = {};
  float stage[16];

  // prefetch chunk 0
#pragma unroll
  for (int j = 0; j < 16; ++j)
    stage[j] = W[(size_t)(skk + 4 * j) * N + n0 + snn];

  for (int c = 0; c < NCHUNK; ++c) {
    __syncthreads();
#pragma unroll
    for (int j = 0; j < 16; ++j)
      sW[(skk + 4 * j) * LDSW + snn] = stage[j];
    __syncthreads();

    if (c + 1 < NCHUNK) {
      const int k0n = (c + 1) * KC;
#pragma unroll
      for (int j = 0; j < 16; ++j)
        stage[j] = W[(size_t)(k0n + skk + 4 * j) * N + n0 + snn];
    }

    const int k0 = c * KC;
#pragma unroll
    for (int kc = 0; kc < KC / 4; ++kc) {
      // A fragment (16x4): lanes 0-15 -> K=+0,+1 ; lanes 16-31 -> K=+2,+3
      const float* ap = &A[(size_t)(m0 + lm) * K + k0 + kc * 4 + lh * 2];
      v2f a = *(const v2f*)ap;  // 8B-aligned b64 load
#pragma unroll
      for (int nt = 0; nt < 4; ++nt) {
        // B fragment (4x16): vgpr0 = rows K0(K1), vgpr1 = rows K2(K3)
        v2f b;
        b.x = sW[(kc * 4 + lh) * LDSW + nt * 16 + lm];
        b.y = sW[(kc * 4 + 2 + lh) * LDSW + nt * 16 + lm];
        acc[nt] = WMMA_F32(a, b, acc[nt]);
      }
    }
  }

  // C layout: vgpr r -> row m0 + r + 8*laneHalf, col n0 + (lane&15)
#pragma unroll
  for (int nt = 0; nt < 4; ++nt) {
    const int col = n0 + nt * 16 + lm;
    const float bv = bias[col];
#pragma unroll
    for (int r = 0; r < 8; ++r) {
      C[(size_t)(m0 + r + 8 * lh) * N + col] = acc[nt][r] + bv;
    }
  }
}

// ---------------------------------------------------------------------------
// Persistent recurrent scan.
//   S holds Xproj on entry (states region of d_out); overwritten in place with
//   h_t = tanh(Xproj_t + h_{t-1} @ W_hh).
// 64 WGs x 128 threads. WG b owns H-columns [16b, 16b+16); its 1024x16 slice
// of W_hh (64 KB) lives in LDS for all 512 steps. Steps separated by a
// device-scope sense-reversing atomic barrier (bar[0]=count, bar[1]=phase).
// Two interleaved WMMA accumulator chains hide accumulation latency; the
// Xproj operands are loaded before the K loop to overlap with the GEMM.
// ---------------------------------------------------------------------------
__global__ __launch_bounds__(128) void rnn_scan_kernel(
    float* __restrict__ S, const float* __restrict__ Whh,
    unsigned* __restrict__ bar) {
  __shared__ float sW[HID * 16];  // 64 KB column slice, stride 16 (conflict-free)

  const int tid = threadIdx.x;
  const int wv = tid >> 5;   // wave -> M sub-tile of B=64
  const int ln = tid & 31;
  const int lh = ln >> 4;
  const int lm = ln & 15;
  const int n0 = blockIdx.x * 16;
  const int m0 = wv * 16;

  // W_hh[:, n0:n0+16] -> LDS, once, reused for all 512 steps
  for (int i = tid; i < HID * 16; i += 128) {
    const int k = i >> 4, n = i & 15;
    sW[i] = Whh[(size_t)k * HID + n0 + n];
  }
  __syncthreads();

  unsigned* cnt = bar;
  unsigned* phs = bar + 1;

  for (int t = 0; t < T_STEPS; ++t) {
    float* st = S + (size_t)t * BATCH * HID;

    // hoist the 8 Xproj operands: independent of the GEMM, latency overlaps
    float xp[8];
#pragma unroll
    for (int r = 0; r < 8; ++r)
      xp[r] = st[(size_t)(m0 + r + 8 * lh) * HID + n0 + lm];

    v8f acc0 = {}, acc1 = {};
    if (t > 0) {
      const float* hprev = S + (size_t)(t - 1) * BATCH * HID;
      const float* ap = &hprev[(size_t)(m0 + lm) * HID + lh * 2];
#pragma unroll 4
      for (int kc = 0; kc < HID / 4; kc += 2) {
        v2f a0 = *(const v2f*)(ap + kc * 4);
        v2f a1 = *(const v2f*)(ap + kc * 4 + 4);
        v2f b0, b1;
        b0.x = sW[(kc * 4 + lh) * 16 + lm];
        b0.y = sW[(kc * 4 + 2 + lh) * 16 + lm];
        b1.x = sW[(kc * 4 + 4 + lh) * 16 + lm];
        b1.y = sW[(kc * 4 + 6 + lh) * 16 + lm];
        acc0 = WMMA_F32(a0, b0, acc0);   // chain 0 (even kc)
        acc1 = WMMA_F32(a1, b1, acc1);   // chain 1 (odd kc)
      }
    }
    v8f acc = acc0 + acc1;

#pragma unroll
    for (int r = 0; r < 8; ++r) {
      const size_t idx = (size_t)(m0 + r + 8 * lh) * HID + n0 + lm;
      st[idx] = TANHF(xp[r] + acc[r]);  // in-place Xproj -> h_t
    }

    if (t < T_STEPS - 1) {
      __threadfence();   // make h_t visible device-wide before arriving
      __syncthreads();
      if (tid == 0) {
        unsigned old = __hip_atomic_fetch_add(cnt, 1u, __ATOMIC_ACQ_REL,
                                              __HIP_MEMORY_SCOPE_AGENT);
        if (old == NWG_SCAN - 1) {
          __hip_atomic_store(cnt, 0u, __ATOMIC_RELAXED, __HIP_MEMORY_SCOPE_AGENT);
          __hip_atomic_fetch_add(phs, 1u, __ATOMIC_RELEASE, __HIP_MEMORY_SCOPE_AGENT);
        } else {
          while (__hip_atomic_load(phs, __ATOMIC_ACQUIRE,
                                   __HIP_MEMORY_SCOPE_AGENT) < (unsigned)(t + 1)) {
            __builtin_amdgcn_s_sleep(2);
          }
        }
      }
      __syncthreads();
      __threadfence();
    }
  }
}

// ---------------------------------------------------------------------------
extern "C" void kernel_launch(void* const* d_in, const int* in_sizes, int n_in,
                              void* d_out, int out_size, void* d_ws, size_t ws_size,
                              hipStream_t stream) {
  const float* X   = (const float*)d_in[0];  // [512,64,512]
  const float* Wxh = (const float*)d_in[1];  // [512,1024]
  const float* Whh = (const float*)d_in[2];  // [1024,1024]
  const float* bh  = (const float*)d_in[3];  // [1024]
  const float* Whq = (const float*)d_in[4];  // [1024,512]
  const float* bq  = (const float*)d_in[5];  // [512]

  float* outputs = (float*)d_out;                                   // T*B*OUTD
  float* states  = outputs + (size_t)T_STEPS * BATCH * OUTD;        // T*B*HID
  unsigned* bar  = (unsigned*)d_ws;

  // barrier state must be zero every call (deterministic under graph replay)
  hipMemsetAsync(bar, 0, 2 * sizeof(unsigned), stream);

  const int M = T_STEPS * BATCH;  // 32768

  // Phase 1: Xproj = X @ W_xh + b_h  -> states region (in-place buffer)
  gemm_bias_kernel<IND, HID>
      <<<dim3(M / 128, HID / 64), dim3(256), 0, stream>>>(X, Wxh, bh, states);

  // Phase 2: persistent recurrent scan, states[t] = h_t
  rnn_scan_kernel<<<dim3(NWG_SCAN), dim3(128), 0, stream>>>(states, Whh, bar);

  // Phase 3: outputs = states @ W_hq + b_q
  gemm_bias_kernel<HID, OUTD>
      <<<dim3(M / 128, OUTD / 64), dim3(256), 0, stream>>>(states, Whq, bq, outputs);
}